// G2_GNN_53712861003962
// MI455X (gfx1250) — compile-verified
//
#include <hip/hip_runtime.h>
#include <hip/hip_bf16.h>

// ---------------------------------------------------------------------------
// G2-GNN forward for MI455X (gfx1250, wave32).
//   - Node GEMMs via V_WMMA_F32_16X16X4_F32 (fp32 matrix pipe, faithful to ref)
//   - Edge GEMM algebraically factored to node GEMMs + per-edge vector ops
//   - Scatter-means via native f32 global atomics (L2-resident tables)
// ---------------------------------------------------------------------------

typedef __attribute__((ext_vector_type(2))) float v2f;
typedef __attribute__((ext_vector_type(8))) float v8f;

#define NFEAT  256
#define NHID   128
#define NCLASS 64

__device__ __forceinline__ void atomAddF(float* p, float v) {
    __hip_atomic_fetch_add(p, v, __ATOMIC_RELAXED, __HIP_MEMORY_SCOPE_AGENT);
}

// ---------------------------------------------------------------------------
// zero fill
__global__ void k_zero(float* __restrict__ p, int n) {
    int i = blockIdx.x * blockDim.x + threadIdx.x;
    if (i < n) p[i] = 0.0f;
}

// ---------------------------------------------------------------------------
// per-node in-degree (dst) and out-degree (src) counts as floats
__global__ void k_count(const int* __restrict__ src, const int* __restrict__ dst,
                        float* __restrict__ cdst, float* __restrict__ csrc, int E) {
    int e = blockIdx.x * blockDim.x + threadIdx.x;
    if (e < E) {
        atomAddF(&cdst[dst[e]], 1.0f);
        atomAddF(&csrc[src[e]], 1.0f);
    }
}

// ---------------------------------------------------------------------------
// out[scat[e], :] += X[gath[e], :]   (NHID=128 wide, 32 threads/edge, float4)
__global__ void k_scatter_rows(const float* __restrict__ X,
                               const int* __restrict__ gath,
                               const int* __restrict__ scat,
                               float* __restrict__ out, int E) {
    int t = blockIdx.x * blockDim.x + threadIdx.x;
    int e = t >> 5;
    if (e >= E) return;
    int f = (t & 31) * 4;
    const float4 v = *(const float4*)(X + (size_t)gath[e] * NHID + f);
    float* o = out + (size_t)scat[e] * NHID + f;
    atomAddF(o + 0, v.x);
    atomAddF(o + 1, v.y);
    atomAddF(o + 2, v.z);
    atomAddF(o + 3, v.w);
}

// ---------------------------------------------------------------------------
// agg[i,:] /= max(cnt[i],1)
__global__ void k_div_rows(float* __restrict__ agg, const float* __restrict__ cnt, int total) {
    int i = blockIdx.x * blockDim.x + threadIdx.x;
    if (i >= total) return;
    float c = cnt[i >> 7];                 // NHID == 128
    agg[i] *= 1.0f / fmaxf(c, 1.0f);
}

// ---------------------------------------------------------------------------
// gate edge work: hs[src[e],:] += (A[src[e],:] + B[dst[e],:])^2   (P = 2.0)
__global__ void k_edge_h(const float* __restrict__ A, const float* __restrict__ B,
                         const int* __restrict__ src, const int* __restrict__ dst,
                         float* __restrict__ hs, int E) {
    int t = blockIdx.x * blockDim.x + threadIdx.x;
    int e = t >> 5;
    if (e >= E) return;
    int f = (t & 31) * 4;
    int s = src[e], d = dst[e];
    const float4 a = *(const float4*)(A + (size_t)s * NHID + f);
    const float4 b = *(const float4*)(B + (size_t)d * NHID + f);
    float hx = (a.x + b.x) * (a.x + b.x);
    float hy = (a.y + b.y) * (a.y + b.y);
    float hz = (a.z + b.z) * (a.z + b.z);
    float hw = (a.w + b.w) * (a.w + b.w);
    float* o = hs + (size_t)s * NHID + f;
    atomAddF(o + 0, hx);
    atomAddF(o + 1, hy);
    atomAddF(o + 2, hz);
    atomAddF(o + 3, hw);
}

// ---------------------------------------------------------------------------
// tau = tanh(hs / max(csrc,1));  Xnext = (1-tau)*Xcur + tau*Xnew
__global__ void k_gate(const float* __restrict__ Xcur, const float* __restrict__ Xnew,
                       const float* __restrict__ hs, const float* __restrict__ csrc,
                       float* __restrict__ Xnext, int total) {
    int i = blockIdx.x * blockDim.x + threadIdx.x;
    if (i >= total) return;
    float c   = fmaxf(csrc[i >> 7], 1.0f);
    float tau = tanhf(hs[i] / c);
    Xnext[i]  = (1.0f - tau) * Xcur[i] + tau * Xnew[i];
}

// ---------------------------------------------------------------------------
// WMMA fp32 GEMM:  C[M,N] = act( A1[M,K] @ W1[K,N] (+ A2[M,K] @ W2[K,N]) + bias )
// One wave -> one 16x16 tile; blockDim 256 (8 waves) -> 16 x (8*16) strip.
// A fragment (16x4 fp32): lane<16 holds K=k0,k0+1 of row lane; lane>=16 K=k0+2,k0+3.
// B fragment (4x16):      lane half selects K pair, lane&15 selects column.
// C/D (16x16 f32, 8 VGPRs): vgpr i -> row i (lanes 0-15) / row i+8 (lanes 16-31).
__global__ void k_gemm(const float* __restrict__ A1, const float* __restrict__ W1,
                       const float* __restrict__ A2, const float* __restrict__ W2,
                       const float* __restrict__ bias, float* __restrict__ C,
                       int M, int K, int N, int relu) {
    const int lane   = threadIdx.x & 31;
    const int wave   = threadIdx.x >> 5;
    const int ntiles = N >> 4;
    if (wave >= ntiles) return;           // whole-wave exit: EXEC stays all-1s for WMMA

    const int m0   = blockIdx.x << 4;
    const int n0   = wave << 4;
    const int row  = lane & 15;
    const int half = lane >> 4;

    v8f c = {};

    {
        const float* arow = A1 + (size_t)(m0 + row) * K;
        for (int k0 = 0; k0 < K; k0 += 4) {
            v2f a, b;
            a.x = arow[k0 + 2 * half];
            a.y = arow[k0 + 2 * half + 1];
            b.x = W1[(size_t)(k0 + 2 * half) * N + n0 + row];
            b.y = W1[(size_t)(k0 + 2 * half + 1) * N + n0 + row];
            c = __builtin_amdgcn_wmma_f32_16x16x4_f32(false, a, false, b,
                                                      (short)0, c, false, false);
        }
    }
    if (A2 != nullptr) {
        const float* arow = A2 + (size_t)(m0 + row) * K;
        for (int k0 = 0; k0 < K; k0 += 4) {
            v2f a, b;
            a.x = arow[k0 + 2 * half];
            a.y = arow[k0 + 2 * half + 1];
            b.x = W2[(size_t)(k0 + 2 * half) * N + n0 + row];
            b.y = W2[(size_t)(k0 + 2 * half + 1) * N + n0 + row];
            c = __builtin_amdgcn_wmma_f32_16x16x4_f32(false, a, false, b,
                                                      (short)0, c, false, false);
        }
    }

    const float bv = bias ? bias[n0 + row] : 0.0f;
    #pragma unroll
    for (int i = 0; i < 8; ++i) {
        float x = c[i] + bv;
        if (relu) x = fmaxf(x, 0.0f);
        int m = m0 + i + 8 * half;
        if (m < M) C[(size_t)m * N + n0 + row] = x;
    }
}

// ---------------------------------------------------------------------------
extern "C" void kernel_launch(void* const* d_in, const int* in_sizes, int n_in,
                              void* d_out, int out_size, void* d_ws, size_t ws_size,
                              hipStream_t stream) {
    const float* X       = (const float*)d_in[0];
    const int*   ei      = (const int*)  d_in[1];
    const float* enc_W   = (const float*)d_in[2];
    const float* enc_b   = (const float*)d_in[3];
    const float* conv_Wl = (const float*)d_in[4];
    const float* conv_bl = (const float*)d_in[5];
    const float* conv_Wr = (const float*)d_in[6];
    const float* gg_Wl   = (const float*)d_in[7];
    const float* gg_bl   = (const float*)d_in[8];
    const float* gg_Wr   = (const float*)d_in[9];
    const float* Q_W     = (const float*)d_in[10];
    const float* Q_b     = (const float*)d_in[11];
    const float* dec_W   = (const float*)d_in[12];
    const float* dec_b   = (const float*)d_in[13];

    const int N = in_sizes[0] / NFEAT;     // 10000 (divisible by 16)
    const int E = in_sizes[1] / 2;         // 320000
    const int* src = ei;
    const int* dst = ei + E;

    // workspace layout (floats)
    float* ws       = (float*)d_ws;
    const size_t NH = (size_t)N * NHID;
    float* cnt_dst  = ws;                  // [N]
    float* cnt_src  = cnt_dst + N;         // [N]
    float* Xa       = cnt_src + N;         // [N,128]
    float* Xb       = Xa + NH;             // [N,128]
    float* agg      = Xb + NH;             // [N,128]
    float* T1       = agg + NH;            // X_  [N,128]
    float* T2       = T1 + NH;             // Xg  [N,128]
    float* Ab       = T2 + NH;             // Xg@Q_top + Q_b
    float* Bb       = Ab + NH;             // Xg@Q_bot
    float* hs       = Bb + NH;             // gate accumulator

    auto cdiv = [](long long a, long long b) { return (int)((a + b - 1) / b); };
    const int total   = (int)NH;
    const int eThr    = E * 32;            // 32 lanes per edge
    const int gemmGrd = N / 16;            // M tiles (10000/16 = 625)

    // degree counts (recomputed each call; graph-capture safe)
    k_zero <<<cdiv(2 * N, 256), 256, 0, stream>>>(cnt_dst, 2 * N);
    k_count<<<cdiv(E, 256),     256, 0, stream>>>(src, dst, cnt_dst, cnt_src, E);

    // encoder: Xa = relu(X @ enc_W + enc_b)
    k_gemm<<<gemmGrd, 256, 0, stream>>>(X, enc_W, nullptr, nullptr, enc_b,
                                        Xa, N, NFEAT, NHID, 1);

    float* Xcur  = Xa;
    float* Xnext = Xb;
    for (int layer = 0; layer < 2; ++layer) {
        // agg = segment_mean(Xcur[src], dst)   (shared by conv and gate paths)
        k_zero        <<<cdiv(total, 256), 256, 0, stream>>>(agg, total);
        k_scatter_rows<<<cdiv(eThr, 256),  256, 0, stream>>>(Xcur, src, dst, agg, E);
        k_div_rows    <<<cdiv(total, 256), 256, 0, stream>>>(agg, cnt_dst, total);

        // X_ = relu(agg@conv_Wl + Xcur@conv_Wr + conv_bl)
        k_gemm<<<gemmGrd, 256, 0, stream>>>(agg, conv_Wl, Xcur, conv_Wr, conv_bl,
                                            T1, N, NHID, NHID, 1);
        // Xg = relu(agg@gg_Wl + Xcur@gg_Wr + gg_bl)
        k_gemm<<<gemmGrd, 256, 0, stream>>>(agg, gg_Wl, Xcur, gg_Wr, gg_bl,
                                            T2, N, NHID, NHID, 1);

        // edge GEMM factored to node level: Ab = Xg@Q_top + Q_b ; Bb = Xg@Q_bot
        k_gemm<<<gemmGrd, 256, 0, stream>>>(T2, Q_W, nullptr, nullptr, Q_b,
                                            Ab, N, NHID, NHID, 0);
        k_gemm<<<gemmGrd, 256, 0, stream>>>(T2, Q_W + (size_t)NHID * NHID, nullptr, nullptr,
                                            nullptr, Bb, N, NHID, NHID, 0);

        // hs[src] += (Ab[src] + Bb[dst])^2 ; then gate
        k_zero  <<<cdiv(total, 256), 256, 0, stream>>>(hs, total);
        k_edge_h<<<cdiv(eThr, 256),  256, 0, stream>>>(Ab, Bb, src, dst, hs, E);
        k_gate  <<<cdiv(total, 256), 256, 0, stream>>>(Xcur, T1, hs, cnt_src, Xnext, total);

        float* t = Xcur; Xcur = Xnext; Xnext = t;
    }

    // decoder: out = Xcur @ dec_W + dec_b
    k_gemm<<<gemmGrd, 256, 0, stream>>>(Xcur, dec_W, nullptr, nullptr, dec_b,
                                        (float*)d_out, N, NHID, NCLASS, 0);
}